// SpanBert_81982335746218
// MI455X (gfx1250) — compile-verified
//
#include <hip/hip_runtime.h>
#include <hip/hip_bf16.h>

typedef __attribute__((ext_vector_type(2))) float v2f;
typedef __attribute__((ext_vector_type(4))) float v4f;
typedef __attribute__((ext_vector_type(8))) float v8f;
typedef __attribute__((ext_vector_type(4))) unsigned int v4u;
typedef __attribute__((ext_vector_type(4))) int v4i;
typedef __attribute__((ext_vector_type(8))) int v8i;

#define Hh 768
#define Bb 4
#define Ss 512
#define Nn 128
#define M_TOTAL (Bb * Nn)      // 512 spans
#define MT (M_TOTAL / 16)      // 32 row strips
#define NT (Hh / 16)           // 48 col tiles
#define WPB 8                  // waves per block (wave32)
#define BLOCK (WPB * 32)
#define NT_PER_WAVE (NT / WPB) // 6 col tiles per wave
#define STRIP_BYTES (16 * Hh * 4)  // 48 KB LDS per A strip

// ---------------------------------------------------------------------------
// TDM descriptor helpers (D# groups per CDNA5 ISA).
// data_size=4B, 2D tensor (tensor_w x tensor_h elems, row stride `stride`),
// tile = tile_w x 16 rows, destination LDS offset `lds_off`.
// 6-arg builtin form (amdgpu-toolchain): (g0 v4u, g1 v8i, v4i, v4i, v8i, cpol)
// ---------------------------------------------------------------------------
__device__ __forceinline__ void tdm_load_strip16(
    unsigned lds_off, const void* gptr,
    unsigned tile_w, unsigned tensor_w, unsigned tensor_h, unsigned stride)
{
    unsigned long long ga = (unsigned long long)gptr;
    v4u g0;
    g0[0] = 1u;                                   // count=1, user mode
    g0[1] = lds_off;                              // lds_addr
    g0[2] = (unsigned)ga;                         // global_addr[31:0]
    g0[3] = ((unsigned)(ga >> 32) & 0x1ffffffu) | (2u << 30);  // addr[56:32], type=2
    v8i g1;
    g1[0] = (int)(2u << 16);                      // data_size=2 (4 bytes)
    g1[1] = (int)(tensor_w << 16);                // tensor_dim0[15:0]
    g1[2] = (int)(tensor_h << 16);                // tensor_dim1[15:0]
    g1[3] = (int)(tile_w << 16);                  // tile_dim0
    g1[4] = 16;                                   // tile_dim1 = 16 rows
    g1[5] = (int)stride;                          // tensor_dim0_stride[31:0]
    g1[6] = 0; g1[7] = 0;
    v4i z4 = {0, 0, 0, 0};
    v8i z8 = {0, 0, 0, 0, 0, 0, 0, 0};
    __builtin_amdgcn_tensor_load_to_lds(g0, g1, z4, z4, z8, 0);
}

// Gather-mode strip load: 16 gathered rows (16-bit indices) of a 2D tensor.
__device__ __forceinline__ void tdm_gather_strip16(
    unsigned lds_off, const void* gptr,
    unsigned tile_w, unsigned tensor_w, unsigned tensor_h, unsigned stride,
    const int* __restrict__ toks)
{
    int gw[8];
    #pragma unroll
    for (int j = 0; j < 8; ++j) {
        const int lo = toks[2 * j], hi = toks[2 * j + 1];
        gw[j] = __builtin_amdgcn_readfirstlane((lo & 0xffff) | (hi << 16));
    }
    unsigned long long ga = (unsigned long long)gptr;
    v4u g0;
    g0[0] = 1u | (0u << 30) | (1u << 31);         // count=1, 16-bit idx, gather_mode
    g0[1] = lds_off;
    g0[2] = (unsigned)ga;
    g0[3] = ((unsigned)(ga >> 32) & 0x1ffffffu) | (2u << 30);
    v8i g1;
    g1[0] = (int)(2u << 16);                      // data_size = 4 bytes
    g1[1] = (int)(tensor_w << 16);
    g1[2] = (int)(tensor_h << 16);
    g1[3] = (int)(tile_w << 16);
    g1[4] = 16;                                   // tile_dim1 = #valid indices
    g1[5] = (int)stride;
    g1[6] = 0; g1[7] = 0;
    v4i g2, g3;
    #pragma unroll
    for (int j = 0; j < 4; ++j) { g2[j] = gw[j]; g3[j] = gw[4 + j]; }
    v8i z8 = {0, 0, 0, 0, 0, 0, 0, 0};
    __builtin_amdgcn_tensor_load_to_lds(g0, g1, g2, g3, z8, 0);
}

// ---------------------------------------------------------------------------
// K1: start_embed / end_embed = gather(seq, idx) @ W + bias
// Block = one 16-row strip; TDM gather stages the 16 gathered seq rows into
// LDS once; each wave computes 6 column tiles with v_wmma_f32_16x16x4_f32.
// ---------------------------------------------------------------------------
__global__ __launch_bounds__(BLOCK) void k_gather_project(
    const float* __restrict__ seq,
    const int* __restrict__ starts, const int* __restrict__ ends,
    const float* __restrict__ Ws, const float* __restrict__ bs,
    const float* __restrict__ We, const float* __restrict__ be,
    float* __restrict__ outS, float* __restrict__ outE)
{
    extern __shared__ float lds[];                // 16 x 768 A strip
    const int lane = threadIdx.x & 31;
    const int wave = threadIdx.x >> 5;
    const int mt   = blockIdx.x;
    const bool isEnd = (blockIdx.y != 0);

    const int*   idxp = isEnd ? ends : starts;
    const float* W    = isEnd ? We   : Ws;
    const float* bias = isEnd ? be   : bs;
    float*       out  = isEnd ? outE : outS;
    const int    bi   = mt >> 3;                  // all 16 rows share one batch

    if (wave == 0) {
        tdm_gather_strip16(0, seq + (size_t)bi * Ss * Hh, Hh, Hh, Ss, Hh,
                           idxp + mt * 16);
        __builtin_amdgcn_s_wait_tensorcnt(0);
    }
    __syncthreads();

    const int khalf = (lane >> 4) * 2;            // lanes 0-15: K 0..1, 16-31: K 2..3
    const int rbase = (lane < 16) ? 0 : 8;
    const float* Alds = lds + (lane & 15) * Hh + khalf;

    for (int t = 0; t < NT_PER_WAVE; ++t) {
        const int nt  = wave * NT_PER_WAVE + t;
        const int col = nt * 16 + (lane & 15);
        const float* Bcol = W + (size_t)khalf * Hh + col;

        v8f acc = {};
        #pragma unroll 4
        for (int k0 = 0; k0 < Hh; k0 += 4) {
            v2f a = *(const v2f*)(Alds + k0);     // ds_load_b64 from staged strip
            v2f b;
            b.x = Bcol[(size_t)k0 * Hh];
            b.y = Bcol[(size_t)(k0 + 1) * Hh];
            acc = __builtin_amdgcn_wmma_f32_16x16x4_f32(false, a, false, b,
                                                        (short)0, acc, false, false);
        }
        const float bv = bias[col];
        #pragma unroll
        for (int r = 0; r < 8; ++r)
            out[(size_t)(mt * 16 + rbase + r) * Hh + col] = acc[r] + bv;
    }
}

// ---------------------------------------------------------------------------
// K2: span_repr = start_embed @ W_out[0:768] + end_embed @ W_out[768:1536]
//               + width_emb[width] @ W_out[1536:1566] + b_out
// Two-phase schedule reusing one 48KB LDS strip: TDM start strip -> pass 1,
// TDM end strip -> pass 2. 6 live accumulators per wave. Width/bias folded
// into accumulator init.
// ---------------------------------------------------------------------------
__global__ __launch_bounds__(BLOCK) void k_span_out(
    const float* __restrict__ startE, const float* __restrict__ endE,
    const int* __restrict__ starts, const int* __restrict__ ends,
    const float* __restrict__ width_emb, const float* __restrict__ W_out,
    const float* __restrict__ b_out, float* __restrict__ spanR)
{
    extern __shared__ float lds[];
    const int lane = threadIdx.x & 31;
    const int wave = threadIdx.x >> 5;
    const int mt   = blockIdx.x;

    if (wave == 0) {
        tdm_load_strip16(0, startE + (size_t)mt * 16 * Hh, Hh, Hh, M_TOTAL, Hh);
        __builtin_amdgcn_s_wait_tensorcnt(0);
    }
    __syncthreads();

    const int khalf = (lane >> 4) * 2;
    const int rbase = (lane < 16) ? 0 : 8;
    const float* Alds = lds + (lane & 15) * Hh + khalf;

    v8f acc[NT_PER_WAVE];

    // Phase 1: width + bias init, then start_embed @ W_out[0:768]
    #pragma unroll
    for (int t = 0; t < NT_PER_WAVE; ++t) {
        const int nt  = wave * NT_PER_WAVE + t;
        const int col = nt * 16 + (lane & 15);
        const float* Bc0 = W_out + col;

        float wcol[30];
        #pragma unroll
        for (int wi = 0; wi < 30; ++wi)
            wcol[wi] = Bc0[(size_t)(2 * Hh + wi) * Hh];

        const float bo = b_out[col];
        #pragma unroll
        for (int r = 0; r < 8; ++r) {
            const int row = mt * 16 + rbase + r;
            int w = ends[row] - starts[row];
            w = w < 0 ? 0 : (w > 10 ? 10 : w);
            float s = bo;
            #pragma unroll
            for (int wi = 0; wi < 30; ++wi)
                s += width_emb[w * 30 + wi] * wcol[wi];
            acc[t][r] = s;
        }

        #pragma unroll 4
        for (int k0 = 0; k0 < Hh; k0 += 4) {
            v2f a = *(const v2f*)(Alds + k0);
            v2f b;
            b.x = Bc0[(size_t)(k0 + khalf) * Hh];
            b.y = Bc0[(size_t)(k0 + khalf + 1) * Hh];
            acc[t] = __builtin_amdgcn_wmma_f32_16x16x4_f32(false, a, false, b,
                                                           (short)0, acc[t], false, false);
        }
    }

    // Swap strip: end_embed
    __syncthreads();
    if (wave == 0) {
        tdm_load_strip16(0, endE + (size_t)mt * 16 * Hh, Hh, Hh, M_TOTAL, Hh);
        __builtin_amdgcn_s_wait_tensorcnt(0);
    }
    __syncthreads();

    // Phase 2: end_embed @ W_out[768:1536], then store
    #pragma unroll
    for (int t = 0; t < NT_PER_WAVE; ++t) {
        const int nt  = wave * NT_PER_WAVE + t;
        const int col = nt * 16 + (lane & 15);
        const float* Bc0 = W_out + col;

        #pragma unroll 4
        for (int k0 = 0; k0 < Hh; k0 += 4) {
            v2f a = *(const v2f*)(Alds + k0);
            v2f b;
            b.x = Bc0[(size_t)(Hh + k0 + khalf) * Hh];
            b.y = Bc0[(size_t)(Hh + k0 + khalf + 1) * Hh];
            acc[t] = __builtin_amdgcn_wmma_f32_16x16x4_f32(false, a, false, b,
                                                           (short)0, acc[t], false, false);
        }
        #pragma unroll
        for (int r = 0; r < 8; ++r)
            spanR[(size_t)(mt * 16 + rbase + r) * Hh + col] = acc[t][r];
    }
}

// ---------------------------------------------------------------------------
// K3: Ai = span_repr @ (Wa + Wc) (y=0), Bj = span_repr @ (Wb - Wc) (y=1)
// span_repr strip staged via TDM; B operand formed on the fly from W1 rows.
// ---------------------------------------------------------------------------
__global__ __launch_bounds__(BLOCK) void k_ai_bj(
    const float* __restrict__ spanR, const float* __restrict__ W1,
    float* __restrict__ Ai, float* __restrict__ Bj)
{
    extern __shared__ float lds[];
    const int lane = threadIdx.x & 31;
    const int wave = threadIdx.x >> 5;
    const int mt   = blockIdx.x;
    const bool isB = (blockIdx.y != 0);

    if (wave == 0) {
        tdm_load_strip16(0, spanR + (size_t)mt * 16 * Hh, Hh, Hh, M_TOTAL, Hh);
        __builtin_amdgcn_s_wait_tensorcnt(0);
    }
    __syncthreads();

    const int khalf = (lane >> 4) * 2;
    const int rbase = (lane < 16) ? 0 : 8;
    const float* Alds = lds + (lane & 15) * Hh + khalf;
    const float  sgn  = isB ? -1.0f : 1.0f;
    float*       out  = isB ? Bj : Ai;

    for (int t = 0; t < NT_PER_WAVE; ++t) {
        const int nt  = wave * NT_PER_WAVE + t;
        const int col = nt * 16 + (lane & 15);
        const float* Wab = W1 + (isB ? (size_t)Hh * Hh : (size_t)0) + col;
        const float* Wc  = W1 + (size_t)2 * Hh * Hh + col;

        v8f acc = {};
        #pragma unroll 4
        for (int k0 = 0; k0 < Hh; k0 += 4) {
            v2f a = *(const v2f*)(Alds + k0);
            v2f b;
            b.x = Wab[(size_t)(k0 + khalf) * Hh] + sgn * Wc[(size_t)(k0 + khalf) * Hh];
            b.y = Wab[(size_t)(k0 + khalf + 1) * Hh] + sgn * Wc[(size_t)(k0 + khalf + 1) * Hh];
            acc = __builtin_amdgcn_wmma_f32_16x16x4_f32(false, a, false, b,
                                                        (short)0, acc, false, false);
        }
        #pragma unroll
        for (int r = 0; r < 8; ++r)
            out[(size_t)(mt * 16 + rbase + r) * Hh + col] = acc[r];
    }
}

// ---------------------------------------------------------------------------
// K4: mention_scores = span_repr @ W_ment + b_ment  (wave per row)
// ---------------------------------------------------------------------------
__global__ __launch_bounds__(BLOCK) void k_mention(
    const float* __restrict__ spanR, const float* __restrict__ W_ment,
    const float* __restrict__ b_ment, float* __restrict__ out)
{
    const int lane = threadIdx.x & 31;
    const int wave = threadIdx.x >> 5;
    const int m    = blockIdx.x * WPB + wave;

    const float* row = spanR + (size_t)m * Hh;
    float s = 0.0f;
    #pragma unroll
    for (int t = 0; t < Hh / 128; ++t) {
        const int h = t * 128 + lane * 4;
        v4f a = *(const v4f*)(row + h);
        v4f w = *(const v4f*)(W_ment + h);
        s += a.x * w.x + a.y * w.y + a.z * w.z + a.w * w.w;
    }
    #pragma unroll
    for (int o = 16; o > 0; o >>= 1) s += __shfl_xor(s, o, 32);
    if (lane == 0) out[m] = s + b_ment[0];
}

// ---------------------------------------------------------------------------
// K5: pw[b,i,j] = sum_h relu(Ai[b,i,h] + Bj[b,j,h] + b1[h]) * W2[h] + b2
// strict lower triangle; zeros elsewhere. One wave per (b,i,j).
// ---------------------------------------------------------------------------
__global__ __launch_bounds__(BLOCK) void k_pairwise(
    const float* __restrict__ Ai, const float* __restrict__ Bj,
    const float* __restrict__ b1, const float* __restrict__ W2,
    const float* __restrict__ b2, float* __restrict__ out)
{
    const int lane = threadIdx.x & 31;
    const int wave = threadIdx.x >> 5;
    const int idx  = blockIdx.x * WPB + wave;   // = b*N*N + i*N + j
    const int bI   = idx >> 14;
    const int i    = (idx >> 7) & 127;
    const int j    = idx & 127;

    if (j >= i) {
        if (lane == 0) out[idx] = 0.0f;
        return;
    }

    const float* ar = Ai + (size_t)(bI * Nn + i) * Hh;
    const float* br = Bj + (size_t)(bI * Nn + j) * Hh;
    float s = 0.0f;
    #pragma unroll
    for (int t = 0; t < Hh / 128; ++t) {
        const int h = t * 128 + lane * 4;
        v4f a = *(const v4f*)(ar + h);
        v4f bb = *(const v4f*)(br + h);
        v4f c = *(const v4f*)(b1 + h);
        v4f w = *(const v4f*)(W2 + h);
        float x;
        x = a.x + bb.x + c.x; s += (x > 0.0f ? x : 0.0f) * w.x;
        x = a.y + bb.y + c.y; s += (x > 0.0f ? x : 0.0f) * w.y;
        x = a.z + bb.z + c.z; s += (x > 0.0f ? x : 0.0f) * w.z;
        x = a.w + bb.w + c.w; s += (x > 0.0f ? x : 0.0f) * w.w;
    }
    #pragma unroll
    for (int o = 16; o > 0; o >>= 1) s += __shfl_xor(s, o, 32);
    if (lane == 0) out[idx] = s + b2[0];
}

// ---------------------------------------------------------------------------
extern "C" void kernel_launch(void* const* d_in, const int* in_sizes, int n_in,
                              void* d_out, int out_size, void* d_ws, size_t ws_size,
                              hipStream_t stream) {
    (void)in_sizes; (void)n_in; (void)out_size; (void)ws_size;

    const float* seq       = (const float*)d_in[0];
    const int*   starts    = (const int*)  d_in[1];
    const int*   ends      = (const int*)  d_in[2];
    const float* W_start   = (const float*)d_in[3];
    const float* b_start   = (const float*)d_in[4];
    const float* W_end     = (const float*)d_in[5];
    const float* b_end     = (const float*)d_in[6];
    const float* width_emb = (const float*)d_in[7];
    const float* W_out     = (const float*)d_in[8];
    const float* b_out     = (const float*)d_in[9];
    const float* W_ment    = (const float*)d_in[10];
    const float* b_ment    = (const float*)d_in[11];
    const float* W1        = (const float*)d_in[12];
    const float* b1        = (const float*)d_in[13];
    const float* W2        = (const float*)d_in[14];
    const float* b2        = (const float*)d_in[15];
    float*       out       = (float*)d_out;

    const size_t MH = (size_t)M_TOTAL * Hh;
    float* ws     = (float*)d_ws;
    float* startE = ws;
    float* endE   = ws + 1 * MH;
    float* spanR  = ws + 2 * MH;
    float* AiW    = ws + 3 * MH;
    float* BjW    = ws + 4 * MH;                // ~7.9 MB workspace

    k_gather_project<<<dim3(MT, 2), BLOCK, STRIP_BYTES, stream>>>(
        seq, starts, ends, W_start, b_start, W_end, b_end, startE, endE);

    k_span_out<<<dim3(MT), BLOCK, STRIP_BYTES, stream>>>(
        startE, endE, starts, ends, width_emb, W_out, b_out, spanR);

    k_ai_bj<<<dim3(MT, 2), BLOCK, STRIP_BYTES, stream>>>(spanR, W1, AiW, BjW);

    k_mention<<<dim3(M_TOTAL / WPB), BLOCK, 0, stream>>>(spanR, W_ment, b_ment, out);

    k_pairwise<<<dim3((Bb * Nn * Nn) / WPB), BLOCK, 0, stream>>>(
        AiW, BjW, b1, W2, b2, out + M_TOTAL);
}